// PNAModule_76235669504169
// MI455X (gfx1250) — compile-verified
//
#include <hip/hip_runtime.h>
#include <hip/hip_bf16.h>

// ---------------------------------------------------------------------------
// PNA (2 layers) for MI455X / gfx1250, fp32 end-to-end using
// V_WMMA_F32_16X16X4_F32 (wave32 fp32 matrix core).
//
// LDS weight layout: K-pair interleaved with pitch 160 words/pair-row so each
// WMMA B operand is a single conflict-free ds_load_b64 into an even VGPR pair.
// ---------------------------------------------------------------------------

typedef __attribute__((ext_vector_type(2))) float v2f;
typedef __attribute__((ext_vector_type(8))) float v8f;

#define F 64
#define WPITCH 160   // words per K-pair row (== 32 mod 64 -> halves in disjoint banks)
#define PPITCH 68    // P-staging row pitch (4*row + kb distinct mod 64)
#define LOG17 2.8332133440562162f

__device__ __forceinline__ v8f wmma4(v2f a, v2f b, v8f c) {
  // D = A(16x4,f32) * B(4x16,f32) + C(16x16,f32)
  return __builtin_amdgcn_wmma_f32_16x16x4_f32(
      false, a, false, b, (short)0, c, false, false);
}

// cooperative global->LDS weight load with K-pair interleave:
//   Wl[(k>>1)*WPITCH + n*2 + (k&1)] = Wg[k*64 + n]
__device__ __forceinline__ void load_w_pairs(const float* __restrict__ Wg,
                                             float* __restrict__ Wl, int K,
                                             int nthreads) {
  for (int idx = threadIdx.x; idx < K * F; idx += nthreads) {
    const int k = idx >> 6;
    const int n = idx & 63;
    Wl[(k >> 1) * WPITCH + n * 2 + (k & 1)] = Wg[idx];
  }
}

// ---------------------------------------------------------------------------
// CSR construction
// ---------------------------------------------------------------------------
__global__ void k_zero_ints(int* __restrict__ p, int n) {
  int i = blockIdx.x * 256 + threadIdx.x;
  if (i < n) p[i] = 0;
}

__global__ void k_count_deg(const int* __restrict__ dst, int* __restrict__ deg, int E) {
  int i = blockIdx.x * 256 + threadIdx.x;
  if (i < E) atomicAdd(&deg[dst[i]], 1);
}

// single-block exclusive scan of deg -> rowptr
__global__ void k_scan_deg(const int* __restrict__ deg, int* __restrict__ rowptr, int N) {
  __shared__ int buf[1024];
  __shared__ int carry;
  if (threadIdx.x == 0) carry = 0;
  __syncthreads();
  for (int base = 0; base < N; base += 1024) {
    int i = base + (int)threadIdx.x;
    int d = (i < N) ? deg[i] : 0;
    buf[threadIdx.x] = d;
    __syncthreads();
    for (int off = 1; off < 1024; off <<= 1) {
      int t = (threadIdx.x >= (unsigned)off) ? buf[threadIdx.x - off] : 0;
      __syncthreads();
      buf[threadIdx.x] += t;
      __syncthreads();
    }
    int incl = buf[threadIdx.x];
    if (i < N) rowptr[i] = carry + incl - d;
    __syncthreads();
    if (threadIdx.x == 1023) carry += buf[1023];
    __syncthreads();
  }
  if (threadIdx.x == 0) rowptr[N] = carry;
}

__global__ void k_copy_ints(const int* __restrict__ a, int* __restrict__ b, int n) {
  int i = blockIdx.x * 256 + threadIdx.x;
  if (i < n) b[i] = a[i];
}

__global__ void k_scatter_edges(const int* __restrict__ dst, int* __restrict__ fill,
                                int* __restrict__ eperm, int E) {
  int i = blockIdx.x * 256 + threadIdx.x;
  if (i < E) {
    int p = atomicAdd(&fill[dst[i]], 1);
    eperm[p] = i;
  }
}

// deterministic order inside each segment (atomics scatter in arbitrary order)
__global__ void k_sort_segments(const int* __restrict__ rowptr, int* __restrict__ eperm, int N) {
  int v = blockIdx.x * 256 + threadIdx.x;
  if (v >= N) return;
  int beg = rowptr[v], end = rowptr[v + 1];
  for (int i = beg + 1; i < end; ++i) {
    int key = eperm[i];
    int j = i - 1;
    while (j >= beg && eperm[j] > key) { eperm[j + 1] = eperm[j]; --j; }
    eperm[j + 1] = key;
  }
}

// ---------------------------------------------------------------------------
// Edge message GEMM: m[e,:] = concat(x[dst[e]], x[src[e]]) @ W_pre + b_pre
// One wave per 16-edge tile, fp32 WMMA 16x16x4, K = 128 (32 chunks).
// ---------------------------------------------------------------------------
__global__ __launch_bounds__(256, 1) void pna_msg(
    const float* __restrict__ xin, const int* __restrict__ src,
    const int* __restrict__ dst, const float* __restrict__ Wpre,
    const float* __restrict__ bpre, float* __restrict__ m, int E) {
  __shared__ float Wl[64 * WPITCH];  // 40 KB, K=128 pair-interleaved
  load_w_pairs(Wpre, Wl, 128, 256);
  __syncthreads();

  const int wave = threadIdx.x >> 5;
  const int lane = threadIdx.x & 31;
  const int half = lane >> 4;   // K sub-pair owned by this lane
  const int row  = lane & 15;   // A-matrix row (M) owned by this lane

  const long tile = (long)blockIdx.x * 8 + wave;
  long e = tile * 16 + row;
  const long ec = (e < E) ? e : (long)(E - 1);
  const int d = dst[ec];
  const int s = src[ec];
  const float* pd = xin + (long)d * F;
  const float* ps = xin + (long)s * F;

  // Preload the whole A stripe for this lane: 32 chunks x float2.
  v2f a[32];
#pragma unroll
  for (int kc = 0; kc < 16; ++kc)
    a[kc] = *(const v2f*)(pd + kc * 4 + half * 2);
#pragma unroll
  for (int kc = 0; kc < 16; ++kc)
    a[16 + kc] = *(const v2f*)(ps + kc * 4 + half * 2);

  v8f acc[4] = {};
#pragma unroll
  for (int kc = 0; kc < 32; ++kc) {
    const int kp = kc * 2 + half;  // K-pair row in LDS
    const float* wrow = Wl + kp * WPITCH;
#pragma unroll
    for (int nt = 0; nt < 4; ++nt) {
      const int n = nt * 16 + row;
      v2f b = *(const v2f*)(wrow + n * 2);  // single ds_load_b64
      acc[nt] = wmma4(a[kc], b, acc[nt]);
    }
  }

#pragma unroll
  for (int nt = 0; nt < 4; ++nt) {
    const float bias = bpre[nt * 16 + row];
#pragma unroll
    for (int r = 0; r < 8; ++r) {
      const long eo = tile * 16 + (r + half * 8);
      if (eo < E) m[eo * F + nt * 16 + row] = acc[nt][r] + bias;
    }
  }
}

// ---------------------------------------------------------------------------
// Segment aggregation over sorted CSR: sum / min / max / std per the reference
// 64 threads per node (one feature each), 4 nodes per 256-thread block.
// ---------------------------------------------------------------------------
__global__ __launch_bounds__(256) void pna_agg(
    const float* __restrict__ m, const int* __restrict__ rowptr,
    const int* __restrict__ eperm, float* __restrict__ agg4, int N) {
  const int v = blockIdx.x * 4 + (threadIdx.x >> 6);
  const int f = threadIdx.x & 63;
  if (v >= N) return;
  const int beg = rowptr[v], end = rowptr[v + 1];
  float s = 0.f, ssq = 0.f;
  float mn = INFINITY, mx = -INFINITY;
  for (int i = beg; i < end; ++i) {
    const int e = eperm[i];
    const float val = m[(long)e * F + f];
    s += val;
    ssq += val * val;
    mn = fminf(mn, val);
    mx = fmaxf(mx, val);
  }
  const int cnt = end - beg;
  if (cnt == 0) { mn = 0.f; mx = 0.f; }
  const float denom = (float)(cnt > 0 ? cnt : 1);
  const float mean = s / denom;
  const float var = ssq / denom - mean * mean;
  const float stdv = sqrtf(fmaxf(var, 0.f) + 1e-5f);
  const long base = (long)v * (4 * F);
  agg4[base + f]           = s;
  agg4[base + F + f]       = mn;
  agg4[base + 2 * F + f]   = mx;
  agg4[base + 3 * F + f]   = stdv;
}

// ---------------------------------------------------------------------------
// Post GEMM: out = ([x | agg | agg*amp | agg/amp] @ W_post + b_post) @ W_lin
//            + b_lin, optional ReLU. One wave per 16-node tile; the 832-wide
//            A-row is generated on the fly (13 panels of 64).
// ---------------------------------------------------------------------------
__global__ __launch_bounds__(128, 1) void pna_post(
    const float* __restrict__ xin, const float* __restrict__ agg4,
    const int* __restrict__ deg, const float* __restrict__ Wpost,
    const float* __restrict__ bpost, const float* __restrict__ Wlin,
    const float* __restrict__ blin, float* __restrict__ out, int N,
    int relu_flag) {
  __shared__ float Wl[32 * WPITCH];        // 20 KB  (W_lin, pair-interleaved)
  __shared__ float Pan[32 * WPITCH];       // 20 KB  (W_post K-panel)
  __shared__ float Pb[4 * 16 * PPITCH];    // 17 KB  (per-wave P staging)

  load_w_pairs(Wlin, Wl, 64, 128);

  const int wave = threadIdx.x >> 5;
  const int lane = threadIdx.x & 31;
  const int half = lane >> 4;
  const int row  = lane & 15;

  const int tile = blockIdx.x * 4 + wave;
  const int v = tile * 16 + row;
  const int vc = (v < N) ? v : (N - 1);
  const int cnt = deg[vc];
  const float degf = (float)(cnt > 0 ? cnt : 1);
  const float amp = logf(degf + 1.0f) * (1.0f / LOG17);
  const float ramp = 1.0f / amp;
  const float* xrow = xin + (long)vc * F;
  const float* arow = agg4 + (long)vc * (4 * F);

  v8f acc[4] = {};
  for (int p = 0; p < 13; ++p) {
    __syncthreads();
    load_w_pairs(Wpost + (long)p * (F * F), Pan, 64, 128);
    if (p < 12)  // hint: pull next panel toward the caches (global_prefetch_b8)
      __builtin_prefetch(Wpost + (long)(p + 1) * (F * F) + threadIdx.x * 32, 0, 0);
    __syncthreads();

    const float* srcp;
    float scale;
    if (p == 0)      { srcp = xrow;                 scale = 1.f;  }
    else if (p < 5)  { srcp = arow + (p - 1) * F;   scale = 1.f;  }
    else if (p < 9)  { srcp = arow + (p - 5) * F;   scale = amp;  }
    else             { srcp = arow + (p - 9) * F;   scale = ramp; }

#pragma unroll
    for (int kc = 0; kc < 16; ++kc) {
      v2f a = *(const v2f*)(srcp + kc * 4 + half * 2);
      a.x *= scale;
      a.y *= scale;
      const float* wrow = Pan + (kc * 2 + half) * WPITCH;
#pragma unroll
      for (int nt = 0; nt < 4; ++nt) {
        v2f b = *(const v2f*)(wrow + (nt * 16 + row) * 2);
        acc[nt] = wmma4(a, b, acc[nt]);
      }
    }
  }

  // P = acc + b_post, staged to LDS (padded pitch) for the second GEMM
  float* Pw = Pb + wave * (16 * PPITCH);
#pragma unroll
  for (int nt = 0; nt < 4; ++nt) {
    const float bias = bpost[nt * 16 + row];
#pragma unroll
    for (int r = 0; r < 8; ++r)
      Pw[(r + half * 8) * PPITCH + nt * 16 + row] = acc[nt][r] + bias;
  }
  __syncthreads();

  v8f acc2[4] = {};
#pragma unroll
  for (int kc = 0; kc < 16; ++kc) {
    v2f a = *(const v2f*)(Pw + row * PPITCH + kc * 4 + half * 2);
    const float* wrow = Wl + (kc * 2 + half) * WPITCH;
#pragma unroll
    for (int nt = 0; nt < 4; ++nt) {
      v2f b = *(const v2f*)(wrow + (nt * 16 + row) * 2);
      acc2[nt] = wmma4(a, b, acc2[nt]);
    }
  }

#pragma unroll
  for (int nt = 0; nt < 4; ++nt) {
    const float bias = blin[nt * 16 + row];
#pragma unroll
    for (int r = 0; r < 8; ++r) {
      const int vo = tile * 16 + (r + half * 8);
      if (vo < N) {
        float val = acc2[nt][r] + bias;
        if (relu_flag) val = fmaxf(val, 0.f);
        out[(long)vo * F + nt * 16 + row] = val;
      }
    }
  }
}

// ---------------------------------------------------------------------------
// Launch
// ---------------------------------------------------------------------------
static inline size_t alignup(size_t x) { return (x + 255) & ~(size_t)255; }

extern "C" void kernel_launch(void* const* d_in, const int* in_sizes, int n_in,
                              void* d_out, int out_size, void* d_ws, size_t ws_size,
                              hipStream_t stream) {
  (void)n_in; (void)out_size; (void)ws_size;
  const int N = in_sizes[0] / F;       // 50000
  const int E = in_sizes[1] / 2;       // 800000

  const float* x   = (const float*)d_in[0];
  const int*   ei  = (const int*)d_in[1];
  const int*   src = ei;
  const int*   dst = ei + E;

  const float* Wpre[2]  = { (const float*)d_in[2],  (const float*)d_in[8]  };
  const float* bpre[2]  = { (const float*)d_in[3],  (const float*)d_in[9]  };
  const float* Wpost[2] = { (const float*)d_in[4],  (const float*)d_in[10] };
  const float* bpost[2] = { (const float*)d_in[5],  (const float*)d_in[11] };
  const float* Wlin[2]  = { (const float*)d_in[6],  (const float*)d_in[12] };
  const float* blin[2]  = { (const float*)d_in[7],  (const float*)d_in[13] };

  // workspace carve-up
  char* ws = (char*)d_ws;
  size_t off = 0;
  int* deg    = (int*)(ws + off); off = alignup(off + (size_t)N * 4);
  int* fill   = (int*)(ws + off); off = alignup(off + (size_t)N * 4);
  int* rowptr = (int*)(ws + off); off = alignup(off + (size_t)(N + 1) * 4);
  int* eperm  = (int*)(ws + off); off = alignup(off + (size_t)E * 4);
  float* m    = (float*)(ws + off); off = alignup(off + (size_t)E * F * 4);
  float* h    = (float*)(ws + off); off = alignup(off + (size_t)N * F * 4);
  float* agg4 = (float*)(ws + off); off = alignup(off + (size_t)N * 4 * F * 4);

  // ---- CSR build (once; both layers share the graph) ----
  k_zero_ints<<<(N + 255) / 256, 256, 0, stream>>>(deg, N);
  k_count_deg<<<(E + 255) / 256, 256, 0, stream>>>(dst, deg, E);
  k_scan_deg<<<1, 1024, 0, stream>>>(deg, rowptr, N);
  k_copy_ints<<<(N + 255) / 256, 256, 0, stream>>>(rowptr, fill, N);
  k_scatter_edges<<<(E + 255) / 256, 256, 0, stream>>>(dst, fill, eperm, E);
  k_sort_segments<<<(N + 255) / 256, 256, 0, stream>>>(rowptr, eperm, N);

  const int etiles = (E + 15) / 16;
  const int ntiles = (N + 15) / 16;
  dim3 msg_grid((etiles + 7) / 8);
  dim3 agg_grid((N + 3) / 4);
  dim3 post_grid((ntiles + 3) / 4);

  // ---- layer 1 ----
  pna_msg<<<msg_grid, 256, 0, stream>>>(x, src, dst, Wpre[0], bpre[0], m, E);
  pna_agg<<<agg_grid, 256, 0, stream>>>(m, rowptr, eperm, agg4, N);
  pna_post<<<post_grid, 128, 0, stream>>>(x, agg4, deg, Wpost[0], bpost[0],
                                          Wlin[0], blin[0], h, N, /*relu=*/1);

  // ---- layer 2 ----
  pna_msg<<<msg_grid, 256, 0, stream>>>(h, src, dst, Wpre[1], bpre[1], m, E);
  pna_agg<<<agg_grid, 256, 0, stream>>>(m, rowptr, eperm, agg4, N);
  pna_post<<<post_grid, 128, 0, stream>>>(h, agg4, deg, Wpost[1], bpost[1],
                                          Wlin[1], blin[1], (float*)d_out, N,
                                          /*relu=*/0);
}